// GIN_2834678415936
// MI455X (gfx1250) — compile-verified
//
#include <hip/hip_runtime.h>
#include <hip/hip_bf16.h>

typedef __attribute__((ext_vector_type(2))) float v2f;
typedef __attribute__((ext_vector_type(8))) float v8f;

#define IN_FEATS  128
#define OUT_FEATS 128

// ---------------------------------------------------------------------------
// Phase 0: zero the aggregation buffer (float4 stores, fully coalesced)
// ---------------------------------------------------------------------------
__global__ void gin_zero_kernel(float4* __restrict__ p, int n4) {
    int i = blockIdx.x * blockDim.x + threadIdx.x;
    if (i < n4) p[i] = make_float4(0.f, 0.f, 0.f, 0.f);
}

// ---------------------------------------------------------------------------
// Phase 1: edge aggregation  agg[dst] += x[src]
// One wave (32 lanes) per edge; each lane owns 4 contiguous features.
// src/dst are wave-uniform -> scalar loads; gather is a b128 load; scatter is
// four global_atomic_add_f32 (no return -> STOREcnt path, RMW at L2).
// ---------------------------------------------------------------------------
__global__ void gin_agg_kernel(const float* __restrict__ x,
                               const long long* __restrict__ src,
                               const long long* __restrict__ dst,
                               float* __restrict__ agg, int n_edges) {
    const int t    = blockIdx.x * blockDim.x + threadIdx.x;
    const int e    = t >> 5;
    const int lane = t & 31;
    if (e >= n_edges) return;

    const long long s = src[e];          // uniform across wave -> s_load
    const long long d = dst[e];

    const float4 v = *(const float4*)(x + s * IN_FEATS + lane * 4);
    float* p = agg + d * IN_FEATS + lane * 4;

    asm volatile("global_atomic_add_f32 %0, %1, off" :: "v"(p + 0), "v"(v.x) : "memory");
    asm volatile("global_atomic_add_f32 %0, %1, off" :: "v"(p + 1), "v"(v.y) : "memory");
    asm volatile("global_atomic_add_f32 %0, %1, off" :: "v"(p + 2), "v"(v.z) : "memory");
    asm volatile("global_atomic_add_f32 %0, %1, off" :: "v"(p + 3), "v"(v.w) : "memory");
}

// ---------------------------------------------------------------------------
// Phase 2: out = (x + agg) @ W^T + b  via V_WMMA_F32_16X16X4_F32 (exact fp32)
// One 16x16 output tile per wave; K loop of 32 WMMAs (K step 4).
//
// Operand layouts (ISA 7.12.2, 32-bit data):
//   A 16x4 : lane l holds A[l%16][kbase + 2*(l>>4) + {0,1}] in 2 VGPRs
//   B 4x16 : lane l holds B[kbase + 2*(l>>4) + {0,1}][l%16]
//            = W[tileN*16 + l%16][kbase + ...]   (row-major W is B-ready)
//   C/D 16x16: VGPR r, lane l -> D[8*(l>>4) + r][l%16]
// ---------------------------------------------------------------------------
__global__ void gin_gemm_kernel(const float* __restrict__ x,
                                const float* __restrict__ agg,
                                const float* __restrict__ W,
                                const float* __restrict__ bias,
                                float* __restrict__ out,
                                int n_tiles_n /* = OUT_FEATS/16 */) {
    const int wave = threadIdx.x >> 5;
    const int lane = threadIdx.x & 31;
    const int tile = blockIdx.x * (blockDim.x >> 5) + wave;
    const int tM   = tile / n_tiles_n;       // 0..624
    const int tN   = tile % n_tiles_n;       // 0..7
    const int half = lane >> 4;              // 0 or 1
    const int l    = lane & 15;

    const int   rowA  = tM * 16 + l;                 // M index this lane serves
    const int   colB  = tN * 16 + l;                 // N index this lane serves
    const float* xr   = x   + (long)rowA * IN_FEATS;
    const float* ar   = agg + (long)rowA * IN_FEATS;
    const float* wr   = W   + (long)colB * IN_FEATS; // W[colB][*]

    v8f acc = {};
    #pragma unroll 4
    for (int k = 0; k < IN_FEATS; k += 4) {
        const int ka = k + half * 2;
        v2f a, bm;
        a.x  = xr[ka]     + ar[ka];
        a.y  = xr[ka + 1] + ar[ka + 1];
        bm.x = wr[ka];
        bm.y = wr[ka + 1];
        acc = __builtin_amdgcn_wmma_f32_16x16x4_f32(
            /*neg_a=*/false, a, /*neg_b=*/false, bm,
            /*c_mod=*/(short)0, acc, /*reuse_a=*/false, /*reuse_b=*/false);
    }

    const int   col = tN * 16 + l;
    const float bv  = bias[col];
    const int   rowBase = tM * 16 + half * 8;
    #pragma unroll
    for (int r = 0; r < 8; ++r) {
        out[(long)(rowBase + r) * OUT_FEATS + col] = acc[r] + bv;
    }
}

// ---------------------------------------------------------------------------
// Launch
// inputs: x [10000*128] f32, src [640000] i64, dst [640000] i64,
//         W [128*128] f32, b [128] f32 ; out [10000*128] f32
// ---------------------------------------------------------------------------
extern "C" void kernel_launch(void* const* d_in, const int* in_sizes, int n_in,
                              void* d_out, int out_size, void* d_ws, size_t ws_size,
                              hipStream_t stream) {
    const float*     x   = (const float*)d_in[0];
    const long long* src = (const long long*)d_in[1];
    const long long* dst = (const long long*)d_in[2];
    const float*     W   = (const float*)d_in[3];
    const float*     b   = (const float*)d_in[4];
    float*           out = (float*)d_out;

    const int n_nodes = in_sizes[0] / IN_FEATS;   // 10000
    const int n_edges = in_sizes[1];              // 640000

    float* agg = (float*)d_ws;                    // n_nodes*IN_FEATS floats (5.12 MB)

    // Phase 0: zero agg
    {
        const int n4 = n_nodes * IN_FEATS / 4;
        const int blk = 256;
        gin_zero_kernel<<<(n4 + blk - 1) / blk, blk, 0, stream>>>((float4*)agg, n4);
    }

    // Phase 1: edge aggregation (wave per edge)
    {
        const long long threads = (long long)n_edges * 32;
        const int blk = 256;
        const int grid = (int)((threads + blk - 1) / blk);
        gin_agg_kernel<<<grid, blk, 0, stream>>>(x, src, dst, agg, n_edges);
    }

    // Phase 2: WMMA GEMM + bias (one 16x16 tile per wave, 8 waves/block)
    {
        const int tilesM = n_nodes / 16;          // 625
        const int tilesN = OUT_FEATS / 16;        // 8
        const int n_tiles = tilesM * tilesN;      // 5000
        const int blk = 256;                      // 8 waves
        const int grid = n_tiles / (blk / 32);    // 625
        gin_gemm_kernel<<<grid, blk, 0, stream>>>(x, agg, W, b, out, tilesN);
    }
}